// Erosion2D_5609227288815
// MI455X (gfx1250) — compile-verified
//
#include <hip/hip_runtime.h>

typedef float v4f __attribute__((ext_vector_type(4)));

#define Bv 32
#define Hv 256
#define Wv 256
#define Cv 4
#define KHv 5
#define KWv 5
#define Fv 8
#define HOv 252
#define WOv 252

#define TILE_W 64                // output columns per block
#define TILE_H 16                // output rows per block
#define IN_W (TILE_W + KWv - 1)  // 68 input pixels wide
#define IN_H (TILE_H + KHv - 1)  // 20 input rows
#define THREADS 256

__global__ __launch_bounds__(THREADS)
__attribute__((amdgpu_waves_per_eu(1, 8)))
void Erosion2D_gfx1250_kernel(const float* __restrict__ x,
                              const float* __restrict__ k,
                              float* __restrict__ out) {
    __shared__ v4f tile[IN_H * IN_W];          // 20*68*16B = 21760 B
    __shared__ v4f ktile[KHv * KWv * Fv];      // [tap][f] -> float4 over c, 3200 B

    const int tid = threadIdx.x;
    const int bx0 = blockIdx.x * TILE_W;
    const int by0 = blockIdx.y * TILE_H;
    const int b   = blockIdx.z;

    // ---- stage kernel weights into LDS, reorganized [tap][f][c] ----
    for (int i = tid; i < KHv * KWv * Fv; i += THREADS) {
        const int f   = i & (Fv - 1);
        const int tap = i >> 3;                 // dy*KW + dx
        v4f kv;
        kv.x = k[(tap * Cv + 0) * Fv + f];
        kv.y = k[(tap * Cv + 1) * Fv + f];
        kv.z = k[(tap * Cv + 2) * Fv + f];
        kv.w = k[(tap * Cv + 3) * Fv + f];
        ktile[i] = kv;
    }

    // ---- async-stage the x tile into LDS (CDNA5 global_load_async_to_lds) ----
    const v4f* __restrict__ xg = (const v4f*)x;   // one pixel (C=4 floats) per v4f
    for (int i = tid; i < IN_H * IN_W; i += THREADS) {
        const int r = i / IN_W;
        const int c = i - r * IN_W;
        int gy = by0 + r; gy = gy > (Hv - 1) ? (Hv - 1) : gy;   // clamped edge reads
        int gx = bx0 + c; gx = gx > (Wv - 1) ? (Wv - 1) : gx;
        const v4f* src = xg + ((size_t)b * Hv * Wv + (size_t)gy * Wv + gx);
        // low 32 bits of a generic LDS pointer == LDS byte offset
        unsigned lds_off = (unsigned)(unsigned long long)&tile[i];
        asm volatile("global_load_async_to_lds_b128 %0, %1, off"
                     :: "v"(lds_off), "v"(src) : "memory");
    }
    asm volatile("s_wait_asynccnt 0" ::: "memory");
    __syncthreads();

    // ---- per-thread compute: 4 adjacent output pixels, filters in 2 phases of 4 ----
    const int px0 = (tid & 15) * 4;   // 0..60 within tile
    const int row = tid >> 4;         // 0..15 within tile
    const int oy  = by0 + row;
    v4f* __restrict__ og = (v4f*)out;

#pragma unroll 1                       // phases sequential: halves live accumulators
    for (int fh = 0; fh < 2; ++fh) {
        float acc[4][4];
#pragma unroll
        for (int i = 0; i < 4; ++i)
#pragma unroll
            for (int fl = 0; fl < 4; ++fl)
                acc[i][fl] = __builtin_inff();

#pragma unroll 1                       // rolled: back-edge fences the scheduler so at
        for (int dy = 0; dy < KHv; ++dy) {   // most 28 ds_loads are in flight -> no spills
            // load 8 consecutive input pixels once, reuse across all dx
            v4f xr[8];
#pragma unroll
            for (int i = 0; i < 8; ++i)
                xr[i] = tile[(row + dy) * IN_W + px0 + i];

#pragma unroll
            for (int dx = 0; dx < KWv; ++dx) {
#pragma unroll
                for (int fl = 0; fl < 4; ++fl) {
                    // all lanes read the same address -> LDS broadcast, no conflicts
                    const v4f kv = ktile[(dy * KWv + dx) * Fv + fh * 4 + fl];
#pragma unroll
                    for (int i = 0; i < 4; ++i) {
                        const v4f xv = xr[dx + i];
                        const float m01 = fminf(xv.x - kv.x, xv.y - kv.y);
                        const float m23 = fminf(xv.z - kv.z, xv.w - kv.w);
                        // min(acc, min(m01, m23)) -> v_min3_num_f32
                        acc[i][fl] = fminf(acc[i][fl], fminf(m01, m23));
                    }
                }
            }
        }

        // ---- streaming (non-temporal) store: 4 filters = one b128 per pixel ----
        if (oy < HOv) {
#pragma unroll
            for (int i = 0; i < 4; ++i) {
                const int ox = bx0 + px0 + i;
                if (ox < WOv) {
                    const size_t pix = ((size_t)b * HOv + oy) * WOv + ox; // pixel = 2 v4f
                    v4f v;
                    v.x = acc[i][0]; v.y = acc[i][1]; v.z = acc[i][2]; v.w = acc[i][3];
                    __builtin_nontemporal_store(v, &og[pix * 2 + fh]);
                }
            }
        }
    }
}

extern "C" void kernel_launch(void* const* d_in, const int* in_sizes, int n_in,
                              void* d_out, int out_size, void* d_ws, size_t ws_size,
                              hipStream_t stream) {
    (void)in_sizes; (void)n_in; (void)d_ws; (void)ws_size; (void)out_size;
    const float* x = (const float*)d_in[0];   // (32, 256, 256, 4) f32
    const float* k = (const float*)d_in[1];   // (5, 5, 4, 8) f32
    float* out = (float*)d_out;               // (32, 252, 252, 8) f32

    dim3 grid((WOv + TILE_W - 1) / TILE_W,    // 4
              (HOv + TILE_H - 1) / TILE_H,    // 16
              Bv);                            // 32
    Erosion2D_gfx1250_kernel<<<grid, THREADS, 0, stream>>>(x, k, out);
}